// HSG_DET_25494925869237
// MI455X (gfx1250) — compile-verified
//
#include <hip/hip_runtime.h>
#include <hip/hip_bf16.h>
#include <math.h>

typedef __attribute__((ext_vector_type(2))) float v2f;
typedef __attribute__((ext_vector_type(8))) float v8f;

#define NG      300
#define NPRED   100000
#define NCLS    80
#define PSTRIDE 85            // 5 + 80
#define GT_TILES 19           // ceil(300/16)
#define GT_PAD   (GT_TILES*16)  // 304
#define PPB      128          // preds per block (8 waves x 16)
#define SIGP     81           // padded class stride (81 coprime with 64 banks)
#define ALPHA_C  0.75f
#define BETA_C   1.0f
#define EPS_C    1e-7f
#define K4PI2    0.40528473456935108577551785283891f  // 4/pi^2
#define KMAXV    10

// ---------------------------------------------------------------------------
// Kernel 1: cost matrix, tiled 16(gt) x 16(pred) per wave via WMMA f32 16x16x4
// ---------------------------------------------------------------------------
__global__ __launch_bounds__(256) void cost_kernel(const float* __restrict__ preds,
                                                   const float* __restrict__ gt_boxes,
                                                   const int*   __restrict__ gt_classes,
                                                   float* __restrict__ cost) {
  __shared__ float s_gx1[GT_PAD], s_gy1[GT_PAD], s_gx2[GT_PAD], s_gy2[GT_PAD];
  __shared__ float s_areag[GT_PAD], s_atg[GT_PAD];
  __shared__ float s_a0l[GT_PAD], s_a0h[GT_PAD], s_a1l[GT_PAD];
  __shared__ int   s_cls[GT_PAD];
  __shared__ float s_raw[NG * 4];     // async-staged gt boxes
  __shared__ int   s_rawc[NG];        // async-staged gt classes
  __shared__ float s_sig[PPB * SIGP]; // per-pred sigmoid(class logits)

  const int tid = threadIdx.x;

  // --- Stage GT data into LDS with CDNA5 async global->LDS copies ----------
  for (int i = tid; i < NG * 4; i += 256) {
    unsigned int lo = (unsigned int)(uintptr_t)(&s_raw[i]);
    const float* ga = gt_boxes + i;
    asm volatile("global_load_async_to_lds_b32 %0, %1, off" :: "v"(lo), "v"(ga) : "memory");
  }
  for (int i = tid; i < NG; i += 256) {
    unsigned int lo = (unsigned int)(uintptr_t)(&s_rawc[i]);
    const int* ga = gt_classes + i;
    asm volatile("global_load_async_to_lds_b32 %0, %1, off" :: "v"(lo), "v"(ga) : "memory");
  }
  asm volatile("s_wait_asynccnt 0x0" ::: "memory");
  __syncthreads();

  // --- Derived per-GT arrays (padded rows get zeros, never stored) ---------
  for (int i = tid; i < GT_PAD; i += 256) {
    float x1 = 0.f, y1 = 0.f, x2 = 0.f, y2 = 0.f; int c = 0;
    if (i < NG) {
      x1 = s_raw[4 * i + 0]; y1 = s_raw[4 * i + 1];
      x2 = s_raw[4 * i + 2]; y2 = s_raw[4 * i + 3];
      c  = s_rawc[i];
    }
    s_gx1[i] = x1; s_gy1[i] = y1; s_gx2[i] = x2; s_gy2[i] = y2;
    s_areag[i] = fmaxf(x2 - x1, 0.f) * fmaxf(y2 - y1, 0.f);
    float wg = fmaxf(x2 - x1, EPS_C), hg = fmaxf(y2 - y1, EPS_C);
    s_atg[i] = atanf(wg / hg);
    float gcx = 0.5f * (x1 + x2), gcy = 0.5f * (y1 + y2);
    s_a0l[i] = -2.f * gcx;            // A row: [-2gcx, -2gcy, gcx^2+gcy^2, 1]
    s_a1l[i] = -2.f * gcy;
    s_a0h[i] = gcx * gcx + gcy * gcy;
    s_cls[i] = c;
  }

  // --- Per-pred sigmoid table (128 preds x 80 cls), hardware exp -----------
  const int pbase = blockIdx.x * PPB;
  for (int i = tid; i < PPB * NCLS; i += 256) {
    int l = i / NCLS, c = i - l * NCLS;
    int pp = min(pbase + l, NPRED - 1);
    float logit = preds[(size_t)pp * PSTRIDE + 5 + c];
    s_sig[l * SIGP + c] = 1.0f / (1.0f + __expf(-logit));
  }

  // --- Per-lane prediction box (lane l and l+16 share a pred) --------------
  const int wave = tid >> 5, lane = tid & 31, ln16 = lane & 15;
  const int lp = wave * 16 + ln16;      // local pred 0..127
  const int p  = pbase + lp;
  const int pc = min(p, NPRED - 1);
  const float xc = preds[(size_t)pc * PSTRIDE + 0];
  const float yc = preds[(size_t)pc * PSTRIDE + 1];
  const float w  = preds[(size_t)pc * PSTRIDE + 2];
  const float h  = preds[(size_t)pc * PSTRIDE + 3];
  const float px1 = xc - 0.5f * w, py1 = yc - 0.5f * h;
  const float px2 = xc + 0.5f * w, py2 = yc + 0.5f * h;
  const float areap = fmaxf(px2 - px1, 0.f) * fmaxf(py2 - py1, 0.f);
  const float atp = atanf(fmaxf(px2 - px1, EPS_C) / fmaxf(py2 - py1, EPS_C));
  const float pcx = 0.5f * (px1 + px2), pcy = 0.5f * (py1 + py2);

  __syncthreads();

  // B (4x16): rows [pcx; pcy; 1; pcx^2+pcy^2]; VGPR0={K0|K2}, VGPR1={K1|K3}
  v2f B;
  B.x = (lane < 16) ? pcx : 1.0f;
  B.y = (lane < 16) ? pcy : (pcx * pcx + pcy * pcy);

  const int halfsel = (lane >> 4) << 3;   // 0 or 8
  for (int gt = 0; gt < GT_TILES; ++gt) {
    const int g0 = gt * 16;
    const int ga = g0 + ln16;
    // A (16x4): lanes 0-15 hold K0,K1; lanes 16-31 hold K2,K3 (same M=ln16)
    v2f A;
    A.x = (lane < 16) ? s_a0l[ga] : s_a0h[ga];
    A.y = (lane < 16) ? s_a1l[ga] : 1.0f;
    v8f acc = {};
    // center_dist2[gt][pred], exact f32 via matrix core
    acc = __builtin_amdgcn_wmma_f32_16x16x4_f32(false, A, false, B,
                                                (short)0, acc, false, false);
#pragma unroll
    for (int r = 0; r < 8; ++r) {
      const int m = g0 + r + halfsel;     // gt row (uniform per half-wave => LDS broadcast)
      const float gx1 = s_gx1[m], gy1 = s_gy1[m];
      const float gx2 = s_gx2[m], gy2 = s_gy2[m];
      const float ix1 = fmaxf(px1, gx1), iy1 = fmaxf(py1, gy1);
      const float ix2 = fminf(px2, gx2), iy2 = fminf(py2, gy2);
      const float inter = fmaxf(ix2 - ix1, 0.f) * fmaxf(iy2 - iy1, 0.f);
      const float uni = areap + s_areag[m] - inter + EPS_C;
      const float iou = inter * __builtin_amdgcn_rcpf(uni);
      const float cw = fmaxf(fmaxf(px2, gx2) - fminf(px1, gx1), EPS_C);
      const float ch = fmaxf(fmaxf(py2, gy2) - fminf(py1, gy1), EPS_C);
      const float c2 = cw * cw + ch * ch + EPS_C;
      const float cd2 = acc[r];
      const float dat = s_atg[m] - atp;
      const float v = K4PI2 * dat * dat;
      const float at = v * __builtin_amdgcn_rcpf(1.0f - iou + v + EPS_C);
      const float ciou = iou - cd2 * __builtin_amdgcn_rcpf(c2) - at * v;
      const float cp = s_sig[lp * SIGP + s_cls[m]];
      const float val = ALPHA_C * (1.0f - cp) +
                        BETA_C * (1.0f - fminf(fmaxf(ciou, 0.f), 1.f));
      if (m < NG && p < NPRED) cost[(size_t)m * NPRED + p] = val;
    }
  }
}

// ---------------------------------------------------------------------------
// Kernel 2: per-GT top-K (smallest cost). Cost matrix (120MB) lives in 192MB L2.
// ---------------------------------------------------------------------------
__global__ __launch_bounds__(256) void topk_kernel(const float* __restrict__ cost,
                                                   float* __restrict__ assign,
                                                   const int* __restrict__ epoch,
                                                   const int* __restrict__ total_epochs) {
  __shared__ unsigned long long s_list[256 * KMAXV];
  __shared__ unsigned long long s_red[256];

  const int tid = threadIdx.x;
  const int g = blockIdx.x;
  const float* row = cost + (size_t)g * NPRED;

  // k_schedule (banker's rounding like Python round())
  const int te = total_epochs[0];
  const float t = (float)epoch[0] / (float)max(1, te - 1);
  int Kt = (int)rintf(10.0f - 9.0f * t);
  Kt = max(1, min(KMAXV, Kt));

  // per-thread sorted ascending top-10 of (orderable_cost_bits << 32 | idx)
  unsigned long long best[KMAXV];
#pragma unroll
  for (int i = 0; i < KMAXV; ++i) best[i] = ~0ull;

  for (int p = tid; p < NPRED; p += 256) {
    unsigned int cb = __float_as_uint(row[p]);
    cb ^= (unsigned int)(((int)cb >> 31)) | 0x80000000u;  // total order for floats
    unsigned long long key = ((unsigned long long)cb << 32) | (unsigned int)p;
    if (key < best[KMAXV - 1]) {
      best[KMAXV - 1] = key;
#pragma unroll
      for (int j = KMAXV - 1; j > 0; --j) {
        unsigned long long a = best[j - 1], b = best[j];
        best[j - 1] = (a < b) ? a : b;
        best[j]     = (a < b) ? b : a;
      }
    }
  }
#pragma unroll
  for (int i = 0; i < KMAXV; ++i) s_list[tid * KMAXV + i] = best[i];
  __syncthreads();

  int head = 0;
  for (int k = 0; k < Kt; ++k) {
    unsigned long long mykey = (head < KMAXV) ? s_list[tid * KMAXV + head] : ~0ull;
    s_red[tid] = mykey;
    __syncthreads();
    for (int s = 128; s > 0; s >>= 1) {
      if (tid < s) {
        unsigned long long a = s_red[tid], b = s_red[tid + s];
        s_red[tid] = (a < b) ? a : b;
      }
      __syncthreads();
    }
    const unsigned long long win = s_red[0];
    __syncthreads();
    if (mykey == win) head++;  // keys unique (contain idx)
    if (tid == 0) {
      const unsigned int pidx = (unsigned int)(win & 0xffffffffu);
      assign[(size_t)(g * Kt + k) * 2 + 0] = (float)pidx;
      assign[(size_t)(g * Kt + k) * 2 + 1] = (float)g;
    }
  }
}

// ---------------------------------------------------------------------------
extern "C" void kernel_launch(void* const* d_in, const int* in_sizes, int n_in,
                              void* d_out, int out_size, void* d_ws, size_t ws_size,
                              hipStream_t stream) {
  (void)in_sizes; (void)n_in; (void)out_size; (void)d_ws; (void)ws_size;
  const float* preds      = (const float*)d_in[0];
  const float* gt_boxes   = (const float*)d_in[1];
  const int*   gt_classes = (const int*)d_in[2];
  const int*   epoch      = (const int*)d_in[3];
  const int*   total_ep   = (const int*)d_in[4];

  float* cost   = (float*)d_out;
  float* assign = cost + (size_t)NG * NPRED;

  const int grid1 = (NPRED + PPB - 1) / PPB;  // 782
  hipLaunchKernelGGL(cost_kernel, dim3(grid1), dim3(256), 0, stream,
                     preds, gt_boxes, gt_classes, cost);
  hipLaunchKernelGGL(topk_kernel, dim3(NG), dim3(256), 0, stream,
                     cost, assign, epoch, total_ep);
}